// TTLinear_8375186227651
// MI455X (gfx1250) — compile-verified
//
#include <hip/hip_runtime.h>

typedef __attribute__((ext_vector_type(2))) float v2f;
typedef __attribute__((ext_vector_type(8))) float v8f;

#define R      16     // TT bond rank carried through the middle
#define KFEAT  4096   // input features
#define NOUT   4096   // output features
#define WAVES  4
#define BLOCK  (WAVES * 32)
#define ROWS_PER_BLOCK (WAVES * 16)
#define KC     512    // K-chunk of A staged in LDS
#define NC     512    // N-chunk of B staged in LDS

// ---------------------------------------------------------------------------
// Build A (4096x16) = f0*f1*f2*f3 and B (16x4096) = f4*f5*f6*f7 in workspace.
// One block; intermediates in LDS. All factors are tiny (<= 2048 floats).
// ---------------------------------------------------------------------------
__global__ __launch_bounds__(256) void tt_build_ab(
    const float* __restrict__ f0, const float* __restrict__ f1,
    const float* __restrict__ f2, const float* __restrict__ f3,
    const float* __restrict__ f4, const float* __restrict__ f5,
    const float* __restrict__ f6, const float* __restrict__ f7,
    float* __restrict__ Ag, float* __restrict__ Bg)
{
    __shared__ float sBig[8192];    // 512x16 (A chain) then 16x512 (B chain)
    __shared__ float sSmall[1024];  // 64x16 (A chain) then 16x64 (B chain)
    const int tid = threadIdx.x;

    // ---- input-side chain: A[(a,b,c,d), v] ----
    // M1[(a,b),t] = sum_s f0[0,a,s] * f1[s,b,t]        (64 x 16)
    for (int i = tid; i < 1024; i += 256) {
        int a = i >> 7, b = (i >> 4) & 7, t = i & 15;
        float acc = 0.f;
        for (int s = 0; s < 16; ++s) acc += f0[a * 16 + s] * f1[(s * 8 + b) * 16 + t];
        sSmall[i] = acc;
    }
    __syncthreads();
    // M2[(ab,c),u] = sum_t M1[ab,t] * f2[t,c,u]        (512 x 16)
    for (int i = tid; i < 8192; i += 256) {
        int ab = i >> 7, c = (i >> 4) & 7, u = i & 15;
        float acc = 0.f;
        for (int t = 0; t < 16; ++t) acc += sSmall[ab * 16 + t] * f2[(t * 8 + c) * 16 + u];
        sBig[i] = acc;
    }
    __syncthreads();
    // A[(abc,d),v] = sum_u M2[abc,u] * f3[u,d,v]       (4096 x 16)
    for (int i = tid; i < 65536; i += 256) {
        int abc = i >> 7, d = (i >> 4) & 7, v = i & 15;
        float acc = 0.f;
        for (int u = 0; u < 16; ++u) acc += sBig[abc * 16 + u] * f3[(u * 8 + d) * 16 + v];
        Ag[i] = acc;
    }
    __syncthreads();

    // ---- output-side chain: B[v, (j,k,l,m)] ----
    // P1[t,(l,m)] = sum_u f6[t,l,u] * f7[u,m,0]        (16 x 64)
    for (int i = tid; i < 1024; i += 256) {
        int t = i >> 6, l = (i >> 3) & 7, m = i & 7;
        float acc = 0.f;
        for (int u = 0; u < 16; ++u) acc += f6[(t * 8 + l) * 16 + u] * f7[u * 8 + m];
        sSmall[i] = acc;
    }
    __syncthreads();
    // P2[s,(k,lm)] = sum_t f5[s,k,t] * P1[t,lm]        (16 x 512)
    for (int i = tid; i < 8192; i += 256) {
        int s = i >> 9, k = (i >> 6) & 7, lm = i & 63;
        float acc = 0.f;
        for (int t = 0; t < 16; ++t) acc += f5[(s * 8 + k) * 16 + t] * sSmall[t * 64 + lm];
        sBig[i] = acc;
    }
    __syncthreads();
    // B[v,(j,klm)] = sum_s f4[v,j,s] * P2[s,klm]       (16 x 4096)
    for (int i = tid; i < 65536; i += 256) {
        int v = i >> 12, j = (i >> 9) & 7, klm = i & 511;
        float acc = 0.f;
        for (int s = 0; s < 16; ++s) acc += f4[(v * 8 + j) * 16 + s] * sBig[s * 512 + klm];
        Bg[i] = acc;
    }
}

// ---------------------------------------------------------------------------
// Fused Y = (X * A) * B + bias using V_WMMA_F32_16X16X4_F32.
// One wave per 16-row tile; 4 waves (64 rows) per block; grid = rows/64.
// ---------------------------------------------------------------------------
__global__ __launch_bounds__(BLOCK) void tt_main(
    const float* __restrict__ X, const float* __restrict__ Ag,
    const float* __restrict__ Bg, const float* __restrict__ bias,
    float* __restrict__ Y)
{
    __shared__ float sTile[8192];        // A chunk (512x16) then B chunk (16x512)
    __shared__ float sV[WAVES * 17 * 16]; // per-wave 16x16 V tile, row stride 17

    const int tid  = threadIdx.x;
    const int wave = tid >> 5;
    const int lane = tid & 31;
    const int lh   = lane >> 4;   // lane half (0/1)
    const int ll   = lane & 15;
    const int rowBase = blockIdx.x * ROWS_PER_BLOCK + wave * 16;
    const float* __restrict__ xrow = X + (size_t)(rowBase + ll) * KFEAT;

    // ---------------- Phase A: V(16x16) = X_tile(16x4096) * A(4096x16) ------
    v8f vacc = {};
    for (int kc = 0; kc < KFEAT; kc += KC) {
        // cooperative stage of A[kc:kc+KC, 0:16] (row-major, 16 floats/row)
        {
            const float4* src = (const float4*)(Ag + (size_t)kc * R);
            float4* dst = (float4*)sTile;
            for (int i = tid; i < (KC * R) / 4; i += BLOCK) dst[i] = src[i];
        }
        __syncthreads();
        for (int k = 0; k < KC; k += 4) {
            // A-fragment of WMMA: X[rowBase+ll, kc+k + 2*lh + {0,1}]
            v2f xa;
            xa.x = xrow[kc + k + 2 * lh];
            xa.y = xrow[kc + k + 2 * lh + 1];
            // B-fragment of WMMA: A[kc+k + 2*lh + {0,1}, ll] from LDS
            v2f aa;
            aa.x = sTile[(k + 2 * lh) * R + ll];
            aa.y = sTile[(k + 2 * lh + 1) * R + ll];
            vacc = __builtin_amdgcn_wmma_f32_16x16x4_f32(
                false, xa, false, aa, (short)0, vacc, false, false);
        }
        __syncthreads();
    }

    // Spill V (C layout: vacc[g] = V[8*lh+g, ll]) and reload as A-fragments.
    {
        float* v = sV + wave * (17 * 16);
        #pragma unroll
        for (int g = 0; g < 8; ++g) v[(lh * 8 + g) * 17 + ll] = vacc[g];
    }
    v2f va[4];
    {
        const float* v = sV + wave * (17 * 16) + ll * 17;
        #pragma unroll
        for (int kk = 0; kk < 4; ++kk) {
            va[kk].x = v[kk * 4 + 2 * lh];
            va[kk].y = v[kk * 4 + 2 * lh + 1];
        }
    }

    // ---------------- Phase B: Y_tile = V(16x16) * B(16x4096) + bias --------
    for (int nc = 0; nc < NOUT; nc += NC) {
        // cooperative stage of B[0:16, nc:nc+NC] (row-major, NC floats/row)
        {
            float4* dst = (float4*)sTile;
            for (int i = tid; i < (16 * NC) / 4; i += BLOCK) {
                int r = i / (NC / 4), c4 = i % (NC / 4);
                dst[i] = *(const float4*)(Bg + (size_t)r * NOUT + nc + c4 * 4);
            }
        }
        __syncthreads();
        for (int n0 = 0; n0 < NC; n0 += 16) {
            v8f yacc = {};
            #pragma unroll
            for (int kk = 0; kk < 4; ++kk) {
                v2f bb;
                bb.x = sTile[(kk * 4 + 2 * lh) * NC + n0 + ll];
                bb.y = sTile[(kk * 4 + 2 * lh + 1) * NC + n0 + ll];
                yacc = __builtin_amdgcn_wmma_f32_16x16x4_f32(
                    false, va[kk], false, bb, (short)0, yacc, false, false);
            }
            float bv = bias[nc + n0 + ll];
            float* yout = Y + (size_t)(rowBase + lh * 8) * NOUT + nc + n0 + ll;
            #pragma unroll
            for (int g = 0; g < 8; ++g) yout[(size_t)g * NOUT] = yacc[g] + bv;
        }
        __syncthreads();
    }
}

extern "C" void kernel_launch(void* const* d_in, const int* in_sizes, int n_in,
                              void* d_out, int out_size, void* d_ws, size_t ws_size,
                              hipStream_t stream) {
    const float* x    = (const float*)d_in[0];
    const float* f0   = (const float*)d_in[1];
    const float* f1   = (const float*)d_in[2];
    const float* f2   = (const float*)d_in[3];
    const float* f3   = (const float*)d_in[4];
    const float* f4   = (const float*)d_in[5];
    const float* f5   = (const float*)d_in[6];
    const float* f6   = (const float*)d_in[7];
    const float* f7   = (const float*)d_in[8];
    const float* bias = (const float*)d_in[9];
    float* out = (float*)d_out;

    float* Ag = (float*)d_ws;            // 4096 x 16
    float* Bg = Ag + (size_t)KFEAT * R;  // 16 x 4096   (total ws use: 512 KB)

    const int rows = in_sizes[0] / KFEAT;          // 8192
    tt_build_ab<<<1, 256, 0, stream>>>(f0, f1, f2, f3, f4, f5, f6, f7, Ag, Bg);
    tt_main<<<rows / ROWS_PER_BLOCK, BLOCK, 0, stream>>>(x, Ag, Bg, bias, out);
}